// GraphLearning_80616536145963
// MI455X (gfx1250) — compile-verified
//
#include <hip/hip_runtime.h>
#include <hip/hip_bf16.h>

typedef __attribute__((ext_vector_type(2))) float v2f;
typedef __attribute__((ext_vector_type(8))) float v8f;

#define Bn 4
#define Nn 512
#define Dn 256
#define Ln 128

// ---------------------------------------------------------------------------
// Kernel 0: Wa[l][d] = W[l][d] * a[l]  (fold adjacency vector into projection),
// plus sum(a) for the mask term.
// ---------------------------------------------------------------------------
__global__ void prep_kernel(const float* __restrict__ W, const float* __restrict__ a,
                            float* __restrict__ Wa, float* __restrict__ sum_a) {
    int t = blockIdx.x * blockDim.x + threadIdx.x;
    const int total = Ln * Dn;
    for (int idx = t; idx < total; idx += gridDim.x * blockDim.x) {
        int l = idx >> 8;          // / Dn
        Wa[idx] = W[idx] * a[l];
    }
    if (blockIdx.x == 0 && threadIdx.x == 0) {
        float s = 0.f;
        for (int l = 0; l < Ln; ++l) s += a[l];
        *sum_a = s;
    }
}

// ---------------------------------------------------------------------------
// Kernel 1: Z[b] = X[b] (512x256) @ Wa^T (256x128), fp32 WMMA 16x16x4.
// One wave per 16x16 C tile; 64 WMMAs along K.
// A 16x4 f32 layout: lanes 0-15 -> K={0,1} in v[0],v[1]; lanes 16-31 -> K={2,3}.
// B 4x16 f32 symmetric (row K striped across lanes, K split by lane half).
// C/D: 8 VGPRs, lane<16 col=lane rows 0..7, lane>=16 col=lane-16 rows 8..15.
// ---------------------------------------------------------------------------
__global__ void __launch_bounds__(128) gemm_wmma_kernel(const float* __restrict__ X,
                                                        const float* __restrict__ Wa,
                                                        float* __restrict__ Z) {
    const int wave = blockIdx.x * (blockDim.x >> 5) + (threadIdx.x >> 5);
    const int lane = threadIdx.x & 31;
    const int b    = wave >> 8;          // 256 tiles per batch (32 M-tiles * 8 L-tiles)
    const int rem  = wave & 255;
    const int i0   = (rem >> 3) << 4;    // M tile origin
    const int l0   = (rem & 7) << 4;     // L tile origin
    const int m    = lane & 15;
    const int kb   = (lane >> 4) << 1;   // 0 or 2

    const float* xrow = X  + ((size_t)(b * Nn + i0 + m) * Dn) + kb;
    const float* wrow = Wa + ((size_t)(l0 + m) * Dn) + kb;

    v8f c = {};
#pragma unroll 4
    for (int k0 = 0; k0 < Dn; k0 += 4) {
        v2f av = *(const v2f*)(xrow + k0);
        v2f bv = *(const v2f*)(wrow + k0);
        c = __builtin_amdgcn_wmma_f32_16x16x4_f32(false, av, false, bv,
                                                  (short)0, c, false, false);
    }

    const int col   = lane & 15;
    const int rbase = (lane >> 4) << 3;  // 0 or 8
    float* zout = Z + ((size_t)(b * Nn + i0 + rbase) * Ln) + l0 + col;
#pragma unroll
    for (int r = 0; r < 8; ++r) zout[(size_t)r * Ln] = c[r];
}

// ---------------------------------------------------------------------------
// Kernel 2: one block per (b,i) row. dist = L1(z_i, z_j) (+ -sum_a if masked),
// leaky_relu, row-max softmax with adj weighting, + 1e-10.
// ---------------------------------------------------------------------------
__global__ void __launch_bounds__(256) dist_softmax_kernel(const float* __restrict__ Z,
                                                           const float* __restrict__ adj,
                                                           const int* __restrict__ box_num,
                                                           const float* __restrict__ sum_a_ptr,
                                                           float* __restrict__ out) {
    const int i = blockIdx.x;
    const int b = blockIdx.y;
    const int t = threadIdx.x;       // 0..255, covers j = t and t+256
    const int lane = t & 31;
    const int wid  = t >> 5;         // 8 waves

    __shared__ float4 zi4[Ln / 4];   // z_i row, 128 floats
    __shared__ float  red[8];

    const float* Zb = Z + (size_t)b * Nn * Ln;
    if (t < Ln / 4) zi4[t] = ((const float4*)(Zb + (size_t)i * Ln))[t];
    __syncthreads();

    const float sum_a = *sum_a_ptr;
    const int   bn    = box_num[b];
    const bool  vi    = i < bn;

    float dd[2];
#pragma unroll
    for (int jj = 0; jj < 2; ++jj) {
        const int j = t + jj * 256;
        const float4* zj = (const float4*)(Zb + (size_t)j * Ln);
        float acc = 0.f;
#pragma unroll 8
        for (int c = 0; c < Ln / 4; ++c) {
            float4 u = zi4[c];
            float4 v = zj[c];
            acc += fabsf(u.x - v.x) + fabsf(u.y - v.y) +
                   fabsf(u.z - v.z) + fabsf(u.w - v.w);
        }
        if (!(vi && (j < bn))) acc -= sum_a;           // mask * sum(a)
        dd[jj] = acc > 0.f ? acc : 0.01f * acc;        // leaky_relu
    }

    // row max (wave32 shuffle + LDS across 8 waves)
    float m = fmaxf(dd[0], dd[1]);
#pragma unroll
    for (int off = 16; off; off >>= 1) m = fmaxf(m, __shfl_xor(m, off, 32));
    if (lane == 0) red[wid] = m;
    __syncthreads();
    float rowmax = red[0];
#pragma unroll
    for (int w = 1; w < 8; ++w) rowmax = fmaxf(rowmax, red[w]);
    __syncthreads();

    const float* adjrow = adj + ((size_t)b * Nn + i) * Nn;
    const float e0 = adjrow[t]       * expf(dd[0] - rowmax);
    const float e1 = adjrow[t + 256] * expf(dd[1] - rowmax);

    float s = e0 + e1;
#pragma unroll
    for (int off = 16; off; off >>= 1) s += __shfl_xor(s, off, 32);
    if (lane == 0) red[wid] = s;
    __syncthreads();
    float rowsum = red[0] + red[1] + red[2] + red[3] +
                   red[4] + red[5] + red[6] + red[7];
    const float inv = 1.0f / rowsum;

    float* orow = out + ((size_t)b * Nn + i) * Nn;
    orow[t]       = e0 * inv + 1e-10f;
    orow[t + 256] = e1 * inv + 1e-10f;
}

// ---------------------------------------------------------------------------
extern "C" void kernel_launch(void* const* d_in, const int* in_sizes, int n_in,
                              void* d_out, int out_size, void* d_ws, size_t ws_size,
                              hipStream_t stream) {
    const float* x       = (const float*)d_in[0];   // (4,512,256)
    const float* adj     = (const float*)d_in[1];   // (4,512,512)
    const int*   box_num = (const int*)  d_in[2];   // (4,1)
    const float* W       = (const float*)d_in[3];   // (128,256)
    const float* a       = (const float*)d_in[4];   // (128,)

    float* outp  = (float*)d_out;                   // (4,512,512)
    float* wsf   = (float*)d_ws;
    float* Wa    = wsf;                             // 128*256 floats
    float* Z     = wsf + Ln * Dn;                   // 4*512*128 floats
    float* sum_a = wsf + Ln * Dn + Bn * Nn * Ln;    // 1 float

    prep_kernel<<<64, 256, 0, stream>>>(W, a, Wa, sum_a);
    gemm_wmma_kernel<<<(Bn * 32 * 8) / 4, 128, 0, stream>>>(x, Wa, Z);
    dist_softmax_kernel<<<dim3(Nn, Bn), 256, 0, stream>>>(Z, adj, box_num, sum_a, outp);
}